// EquivariantLayerBlock_10934986735881
// MI455X (gfx1250) — compile-verified
//
#include <hip/hip_runtime.h>

typedef __attribute__((ext_vector_type(2))) float v2f;
typedef __attribute__((ext_vector_type(8))) float v8f;

#define N0 1024
#define N1 512
#define JSTRIDE ((size_t)N1 * (size_t)N0)  /* 524288: stride of first index  */
#define KSTRIDE ((size_t)N0)               /* 1024:   stride of second index */

// out[j*1024 + a] = sum_{t=0..511} X[j*FIX + t*LOOP + a]
// One wave handles 16 consecutive 'a' values for one 'j'.
// WMMA trick: D += ones(16x4) x B(4x16)  =>  every row of D accumulates the
// column sums of B over its 4 K-slices; looped over the 512-long axis this
// leaves the full reduction in row M=0 of the accumulator.
// B layout (4x16 f32, 2 VGPRs, wave32): lane n<16 -> N=n, K={0,1} in v0/v1;
// lane n>=16 -> N=n-16, K={2,3} in v0/v1 (mirrors the documented A 16x4 layout).
// FIX/LOOP are compile-time so the unrolled loads use immediate offsets.
template <size_t FIX, size_t LOOP>
__global__ __launch_bounds__(256) void sum512_wmma(const float* __restrict__ X,
                                                   float* __restrict__ out) {
  const int lane  = threadIdx.x & 31;
  const int wv    = threadIdx.x >> 5;
  const int j     = blockIdx.y;
  const int a0    = (blockIdx.x * 8 + wv) * 16;
  const int an    = lane & 15;
  const int khalf = (lane >> 4) << 1;  // 0 or 2

  const float* base = X + (size_t)j * FIX + (size_t)khalf * LOOP
                        + (size_t)(a0 + an);

  const v2f ones = {1.0f, 1.0f};
  v8f acc = {};

#pragma unroll 4
  for (int t = 0; t < 512; t += 4) {
    v2f b;
    b.x = base[(size_t)t * LOOP];
    b.y = base[(size_t)t * LOOP + LOOP];
    acc = __builtin_amdgcn_wmma_f32_16x16x4_f32(false, ones, false, b,
                                                (short)0, acc, false, false);
  }

  // Row M=0 of D: VGPR0 (acc[0]), lanes 0..15, N = lane.
  if (lane < 16) out[(size_t)j * N0 + (size_t)(a0 + lane)] = acc[0];
}

__global__ __launch_bounds__(256) void diag_gather(const float* __restrict__ X,
                                                   float* __restrict__ Dg) {
  const int a = blockIdx.x * 256 + threadIdx.x;
  const int j = blockIdx.y;
  Dg[(size_t)j * N0 + a] = X[(size_t)j * JSTRIDE + (size_t)j * KSTRIDE + a];
}

// T1[a] = sum_j R[j,a] ; T2[a] = sum_j Diag[j,a]
__global__ __launch_bounds__(256) void reduce_over_j(const float* __restrict__ R,
                                                     const float* __restrict__ Dg,
                                                     float* __restrict__ T1,
                                                     float* __restrict__ T2) {
  const int a = blockIdx.x * 256 + threadIdx.x;
  float t1 = 0.0f, t2 = 0.0f;
  for (int j = 0; j < N1; ++j) {
    t1 += R[(size_t)j * N0 + a];
    t2 += Dg[(size_t)j * N0 + a];
  }
  T1[a] = t1;
  T2[a] = t2;
}

// r[j] = sum_a R[j,a] ; c[j] = sum_a C[j,a] ; d[j] = sum_a Diag[j,a]
__global__ __launch_bounds__(256) void reduce_over_a(const float* __restrict__ R,
                                                     const float* __restrict__ C,
                                                     const float* __restrict__ Dg,
                                                     float* __restrict__ r,
                                                     float* __restrict__ c,
                                                     float* __restrict__ d) {
  __shared__ float s0[256], s1[256], s2[256];
  const int j = blockIdx.x;
  const int t = threadIdx.x;
  float pr = 0.0f, pc = 0.0f, pd = 0.0f;
  for (int a = t; a < N0; a += 256) {
    pr += R[(size_t)j * N0 + a];
    pc += C[(size_t)j * N0 + a];
    pd += Dg[(size_t)j * N0 + a];
  }
  s0[t] = pr; s1[t] = pc; s2[t] = pd;
  __syncthreads();
  for (int off = 128; off > 0; off >>= 1) {
    if (t < off) {
      s0[t] += s0[t + off];
      s1[t] += s1[t + off];
      s2[t] += s2[t + off];
    }
    __syncthreads();
  }
  if (t == 0) {
    r[j] = s0[0];
    c[j] = s1[0];
    d[j] = s2[0];
  }
}

// s[0] = sum_a T1[a] ; s[1] = sum_a T2[a]
__global__ __launch_bounds__(256) void reduce_scalars(const float* __restrict__ T1,
                                                      const float* __restrict__ T2,
                                                      float* __restrict__ s) {
  __shared__ float sh1[256], sh2[256];
  const int t = threadIdx.x;
  float p1 = 0.0f, p2 = 0.0f;
  for (int a = t; a < N0; a += 256) {
    p1 += T1[a];
    p2 += T2[a];
  }
  sh1[t] = p1; sh2[t] = p2;
  __syncthreads();
  for (int off = 128; off > 0; off >>= 1) {
    if (t < off) {
      sh1[t] += sh1[t + off];
      sh2[t] += sh2[t + off];
    }
    __syncthreads();
  }
  if (t == 0) {
    s[0] = sh1[0];
    s[1] = sh2[0];
  }
}

// Y[a,j] = w0*T1[a] + w1*T2[a] + w2*R[j,a] + w3*C[j,a] + w4*Diag[j,a]
//        + w5*s1 + w6*s2 + w7*r[j] + w8*c[j] + w9*d[j]
// 16x16 LDS transpose so both the [j][a] reads and the [a][j] writes coalesce.
__global__ __launch_bounds__(256) void finalize(const float* __restrict__ R,
                                                const float* __restrict__ C,
                                                const float* __restrict__ Dg,
                                                const float* __restrict__ T1,
                                                const float* __restrict__ T2,
                                                const float* __restrict__ r,
                                                const float* __restrict__ c,
                                                const float* __restrict__ d,
                                                const float* __restrict__ s,
                                                const float* __restrict__ w,
                                                float* __restrict__ Y) {
  __shared__ float sR[16][17], sC[16][17], sD[16][17];
  const int tx = threadIdx.x, ty = threadIdx.y;
  const int a0 = blockIdx.x * 16, j0 = blockIdx.y * 16;
  const size_t src = (size_t)(j0 + ty) * N0 + (size_t)(a0 + tx);
  sR[ty][tx] = R[src];
  sC[ty][tx] = C[src];
  sD[ty][tx] = Dg[src];
  __syncthreads();
  const int a = a0 + ty;
  const int jj = j0 + tx;
  const float y = w[0] * T1[a] + w[1] * T2[a]
                + w[2] * sR[tx][ty] + w[3] * sC[tx][ty] + w[4] * sD[tx][ty]
                + w[5] * s[0] + w[6] * s[1]
                + w[7] * r[jj] + w[8] * c[jj] + w[9] * d[jj];
  Y[(size_t)a * N1 + jj] = y;
}

extern "C" void kernel_launch(void* const* d_in, const int* in_sizes, int n_in,
                              void* d_out, int out_size, void* d_ws, size_t ws_size,
                              hipStream_t stream) {
  const float* X = (const float*)d_in[0];
  const float* w = (const float*)d_in[1];
  float* Y = (float*)d_out;

  float* ws = (float*)d_ws;
  float* R  = ws;                  // 512*1024
  float* C  = ws + 524288;         // 512*1024
  float* Dg = ws + 1048576;        // 512*1024
  float* T1 = ws + 1572864;        // 1024
  float* T2 = T1 + 1024;           // 1024
  float* rj = T2 + 1024;           // 512
  float* cj = rj + 512;            // 512
  float* dj = cj + 512;            // 512
  float* s  = dj + 512;            // 2

  const dim3 b256(256);

  // Pass 1: R[j,a] = sum_k X[j,k,a]   (reads 1 GiB, WMMA reducer)
  sum512_wmma<JSTRIDE, KSTRIDE><<<dim3(8, N1), b256, 0, stream>>>(X, R);
  // Pass 2: C[j,a] = sum_i X[i,j,a]   (reads 1 GiB, WMMA reducer)
  sum512_wmma<KSTRIDE, JSTRIDE><<<dim3(8, N1), b256, 0, stream>>>(X, C);
  // Diagonal gather: Dg[j,a] = X[j,j,a]
  diag_gather<<<dim3(4, N1), b256, 0, stream>>>(X, Dg);
  // Per-a reductions over j
  reduce_over_j<<<dim3(4), b256, 0, stream>>>(R, Dg, T1, T2);
  // Per-j reductions over a
  reduce_over_a<<<dim3(N1), b256, 0, stream>>>(R, C, Dg, rj, cj, dj);
  // Scalars
  reduce_scalars<<<dim3(1), b256, 0, stream>>>(T1, T2, s);
  // Combine into Y
  finalize<<<dim3(N0 / 16, N1 / 16), dim3(16, 16), 0, stream>>>(
      R, C, Dg, T1, T2, rj, cj, dj, s, w, Y);
}